// DetectionLoss_13048110645457
// MI455X (gfx1250) — compile-verified
//
#include <hip/hip_runtime.h>
#include <hip/hip_bf16.h>
#include <math.h>

typedef float v2f  __attribute__((ext_vector_type(2)));
typedef float v8f  __attribute__((ext_vector_type(8)));
typedef unsigned int u32x4 __attribute__((ext_vector_type(4)));
typedef int   i32x4 __attribute__((ext_vector_type(4)));
typedef int   i32x8 __attribute__((ext_vector_type(8)));

#define EPS_F 1e-7f
#define MAXG 64

__device__ __forceinline__ float softplusf(float x) {
    // numerically stable log(1+exp(x))
    return fmaxf(x, 0.0f) + log1pf(expf(-fabsf(x)));
}

// Exact fp32 sum of one value per lane across the whole wave32, using
// V_WMMA_F32_16X16X4_F32.  Stage 1: A holds the lane value duplicated in
// both A-VGPRs (A[m,0..1]=v_m, A[m,2..3]=v_{m+16}), B = ones (layout
// invariant) => D1[m,n] = 2*(v_m + v_{m+16}), constant across columns.
// Stage 2: A = ones (layout invariant), B fed from D1 VGPR pairs; since
// D1 rows are column-constant, each chained WMMA adds 4 distinct rows of
// D1 exactly, independent of the B (k,n) lane mapping.  After 4 chained
// WMMAs every element of acc == 2 * (sum over 32 lanes).
__device__ __forceinline__ float wave_reduce_wmma(float v) {
    v2f a;   a[0] = v;    a[1] = v;
    v2f one; one[0] = 1.0f; one[1] = 1.0f;
    v8f z   = {0.f,0.f,0.f,0.f,0.f,0.f,0.f,0.f};
    v8f d1 = __builtin_amdgcn_wmma_f32_16x16x4_f32(false, a, false, one, (short)0, z, false, false);
    v8f acc = {0.f,0.f,0.f,0.f,0.f,0.f,0.f,0.f};
    v2f b;
    b[0] = d1[0]; b[1] = d1[1];
    acc = __builtin_amdgcn_wmma_f32_16x16x4_f32(false, one, false, b, (short)0, acc, false, false);
    b[0] = d1[2]; b[1] = d1[3];
    acc = __builtin_amdgcn_wmma_f32_16x16x4_f32(false, one, false, b, (short)0, acc, false, false);
    b[0] = d1[4]; b[1] = d1[5];
    acc = __builtin_amdgcn_wmma_f32_16x16x4_f32(false, one, false, b, (short)0, acc, false, false);
    b[0] = d1[6]; b[1] = d1[7];
    acc = __builtin_amdgcn_wmma_f32_16x16x4_f32(false, one, false, b, (short)0, acc, false, false);
    return acc[0] * 0.5f;
}

// -------------------- kernel 1: sum softplus(pred_cls) --------------------
__global__ void k_cls_softplus(const float* __restrict__ cls, long long n,
                               float* __restrict__ partsA) {
    __shared__ float sred[8];
    long long gid    = (long long)blockIdx.x * blockDim.x + threadIdx.x;
    long long stride = (long long)gridDim.x * blockDim.x;
    long long n4 = n >> 2;
    const float4* c4 = (const float4*)cls;
    float s = 0.0f;
    for (long long k = gid; k < n4; k += stride) {
        float4 v = c4[k];             // global_load_b128, fully coalesced
        s += softplusf(v.x) + softplusf(v.y) + softplusf(v.z) + softplusf(v.w);
    }
    if (gid == 0)
        for (long long k = (n4 << 2); k < n; ++k) s += softplusf(cls[k]);
    float w = wave_reduce_wmma(s);
    int wave = threadIdx.x >> 5, lane = threadIdx.x & 31;
    if (lane == 0) sred[wave] = w;
    __syncthreads();
    if (threadIdx.x == 0) {
        float t = 0.0f;
        for (int q = 0; q < 8; ++q) t += sred[q];   // fixed order: deterministic
        partsA[blockIdx.x] = t;
    }
}

// --------- kernel 2: assignment + CIoU + obj BCE + cls label term ---------
__global__ void __launch_bounds__(256)
k_assign(const float* __restrict__ cls, const float* __restrict__ obj,
         const float* __restrict__ dec, const float* __restrict__ points,
         const float* __restrict__ strides, const float* __restrict__ gtb,
         const int* __restrict__ gtl, const unsigned char* __restrict__ gtv,
         int P, int G, int C, float* __restrict__ partsB, int NB) {
    __shared__ float sB[MAXG * 4];
    __shared__ int   sL[MAXG];
    __shared__ float sV[MAXG];
    __shared__ float sred[8 * 4];

    const int b = blockIdx.y;

    // --- TDM: DMA the 4*G-float GT box tile for batch b into LDS -----------
    if (threadIdx.x < 32) {                   // wave 0 only (wave-uniform branch)
        unsigned lds_addr = (unsigned)(size_t)(void*)sB;      // low 32 bits = LDS byte offset
        unsigned long long ga = (unsigned long long)(size_t)(gtb + (size_t)b * G * 4);
        unsigned n0 = (unsigned)(4 * G);      // elements (data_size = 4B)
        u32x4 g0;
        g0[0] = 1u;                                            // count=1, user D#
        g0[1] = lds_addr;                                      // lds_addr[31:0]
        g0[2] = (unsigned)(ga & 0xffffffffu);                  // global_addr[31:0]
        g0[3] = (unsigned)((ga >> 32) & 0x01ffffffu) | (2u << 30); // addr[56:32] | type=2
        i32x8 g1;
        g1[0] = 0x00020000;         // workgroup_mask=0, data_size=2 (4B)
        g1[1] = (int)(n0 << 16);    // tensor_dim0[15:0] in [31:16]
        g1[2] = (int)(1u << 16);    // tensor_dim0 hi = 0 ; tensor_dim1 lo = 1
        g1[3] = (int)(n0 << 16);    // tensor_dim1 hi = 0 ; tile_dim0 = n0
        g1[4] = 1;                  // tile_dim1 = 1, tile_dim2 = 0
        g1[5] = (int)n0;            // tensor_dim0_stride lo
        g1[6] = 0;                  // stride hi / dim1_stride lo
        g1[7] = 0;
        i32x4 g2 = {0, 0, 0, 0};
        i32x4 g3 = {0, 0, 0, 0};
#if defined(__clang_major__) && __clang_major__ >= 23
        i32x8 g4 = {0, 0, 0, 0, 0, 0, 0, 0};
        __builtin_amdgcn_tensor_load_to_lds(g0, g1, g2, g3, g4, 0);
#else
        __builtin_amdgcn_tensor_load_to_lds(g0, g1, g2, g3, 0);
#endif
        __builtin_amdgcn_s_wait_tensorcnt(0);
    }
    if ((int)threadIdx.x < G) {
        sL[threadIdx.x] = gtl[b * G + threadIdx.x];
        sV[threadIdx.x] = gtv[b * G + threadIdx.x] ? 1.0f : 0.0f;
    }
    __syncthreads();

    const int p   = blockIdx.x * blockDim.x + threadIdx.x;
    const int act = (p < P);
    const int pc  = act ? p : (P - 1);

    const float px  = points[2 * pc];
    const float py  = points[2 * pc + 1];
    const float st  = strides[pc];
    const float r8  = st * 8.0f;
    const float rad = st * 2.5f;

    float bestAm = INFINITY, bestAf = INFINITY;
    int   idxm = 0, idxf = 0, anym = 0, anyf = 0;
    for (int g = 0; g < G; ++g) {
        float x1 = sB[4*g+0], y1 = sB[4*g+1], x2 = sB[4*g+2], y2 = sB[4*g+3];
        float l = px - x1, t = py - y1, r = x2 - px, bo = y2 - py;
        float mn = fminf(fminf(l, t), fminf(r, bo));
        float mx = fmaxf(fmaxf(l, t), fmaxf(r, bo));
        float cx = (x1 + x2) * 0.5f, cy = (y1 + y2) * 0.5f;
        float cmn = fminf(fminf(px - (cx - rad), py - (cy - rad)),
                          fminf(cx + rad - px,  cy + rad - py));
        int base = (mn > 0.0f) && (cmn > 0.0f) && (sV[g] > 0.0f);
        int matc = base && (mx <= r8);
        float area = (x2 - x1) * (y2 - y1);
        if (matc && area < bestAm) { bestAm = area; idxm = g; }
        anym |= matc;
        if (base && area < bestAf) { bestAf = area; idxf = g; }
        anyf |= base;
    }
    const int   fg  = act && (anym || anyf);
    const int   idx = anym ? idxm : idxf;
    const float fgf = fg ? 1.0f : 0.0f;

    float tx1 = fg ? sB[4*idx+0] : 0.0f, ty1 = fg ? sB[4*idx+1] : 0.0f;
    float tx2 = fg ? sB[4*idx+2] : 0.0f, ty2 = fg ? sB[4*idx+3] : 0.0f;

    const float4 pb = ((const float4*)dec)[(size_t)b * P + pc];
    float px1 = pb.x, py1 = pb.y, px2 = pb.z, py2 = pb.w;

    float iw = fmaxf(fminf(px2, tx2) - fmaxf(px1, tx1), 0.0f);
    float ih = fmaxf(fminf(py2, ty2) - fmaxf(py1, ty1), 0.0f);
    float inter = iw * ih;
    float uni = (px2 - px1) * (py2 - py1) + (tx2 - tx1) * (ty2 - ty1) - inter;
    float iou = inter / (uni + EPS_F);
    float cw = fmaxf(px2, tx2) - fminf(px1, tx1);
    float ch = fmaxf(py2, ty2) - fminf(py1, ty1);
    float diag = cw * cw + ch * ch + EPS_F;
    float dx = (px1 + px2 - tx1 - tx2), dy = (py1 + py2 - ty1 - ty2);
    float dist = (dx * dx + dy * dy) * 0.25f;
    float at = atanf((tx2 - tx1) / (ty2 - ty1 + EPS_F))
             - atanf((px2 - px1) / (py2 - py1 + EPS_F));
    float vv = 0.405284734569351f * at * at;              // 4/pi^2
    float alpha = vv / (1.0f - iou + vv + EPS_F);
    float ciou = 1.0f - iou + dist / diag + alpha * vv;
    if (!isfinite(ciou)) ciou = 1.0f;                     // nan_to_num(1,1,1)

    float tgt  = fminf(fmaxf(1.0f - ciou, 0.0f), 1.0f) * fgf;
    float o    = obj[(size_t)b * P + pc];
    float a_obj = act ? (softplusf(o) - o * tgt) : 0.0f;
    float xlab  = fg ? cls[((size_t)b * P + pc) * C + sL[idx]] : 0.0f;
    float a_cls = -fgf * xlab;
    float a_box = ciou * fgf;
    float a_nfg = fgf;

    // --- block reduction (WMMA wave reduce + fixed-order LDS combine) ------
    float w0 = wave_reduce_wmma(a_nfg);
    float w1 = wave_reduce_wmma(a_box);
    float w2 = wave_reduce_wmma(a_cls);
    float w3 = wave_reduce_wmma(a_obj);
    int wave = threadIdx.x >> 5, lane = threadIdx.x & 31;
    if (lane == 0) {
        sred[wave*4+0] = w0; sred[wave*4+1] = w1;
        sred[wave*4+2] = w2; sred[wave*4+3] = w3;
    }
    __syncthreads();
    if (threadIdx.x == 0) {
        float t0 = 0, t1 = 0, t2 = 0, t3 = 0;
        for (int q = 0; q < 8; ++q) {
            t0 += sred[q*4+0]; t1 += sred[q*4+1];
            t2 += sred[q*4+2]; t3 += sred[q*4+3];
        }
        int blk = blockIdx.y * gridDim.x + blockIdx.x;
        partsB[0*NB + blk] = t0;
        partsB[1*NB + blk] = t1;
        partsB[2*NB + blk] = t2;
        partsB[3*NB + blk] = t3;
    }
}

// -------------------- kernel 3: deterministic finalize --------------------
__device__ __forceinline__ float block_reduce_256(float v, float* sred) {
    float w = wave_reduce_wmma(v);
    int wave = threadIdx.x >> 5, lane = threadIdx.x & 31;
    __syncthreads();
    if (lane == 0) sred[wave] = w;
    __syncthreads();
    float r = 0.0f;
    for (int q = 0; q < 8; ++q) r += sred[q];   // same value in every thread
    return r;
}

__global__ void __launch_bounds__(256)
k_final(const float* __restrict__ partsA, int NA,
        const float* __restrict__ partsB, int NB, float* __restrict__ out) {
    __shared__ float sred[8];
    float tA = 0.0f;
    for (int i = threadIdx.x; i < NA; i += 256) tA += partsA[i];
    float vA = block_reduce_256(tA, sred);
    float vs[4];
    for (int j = 0; j < 4; ++j) {
        float t = 0.0f;
        for (int i = threadIdx.x; i < NB; i += 256) t += partsB[j * NB + i];
        vs[j] = block_reduce_256(t, sred);
    }
    if (threadIdx.x == 0) {
        float nfg  = vs[0];
        float norm = fmaxf(nfg, 1.0f);
        float lbox = vs[1] / norm;
        float lcls = (vA + vs[2]) / norm;   // softplus sum + (-sum fg*x_label)
        float lobj = vs[3] / norm;
        float total = 1.0f * lcls + 5.0f * lbox + 1.0f * lobj;
        out[0] = total; out[1] = lcls; out[2] = lbox; out[3] = lobj; out[4] = nfg;
    }
}

extern "C" void kernel_launch(void* const* d_in, const int* in_sizes, int n_in,
                              void* d_out, int out_size, void* d_ws, size_t ws_size,
                              hipStream_t stream) {
    const float* pred_cls = (const float*)d_in[0];
    // d_in[1] = pred_box: unused by the reference loss
    const float* pred_obj = (const float*)d_in[2];
    const float* dec      = (const float*)d_in[3];
    const float* points   = (const float*)d_in[4];
    const float* strides  = (const float*)d_in[5];
    const float* gtb      = (const float*)d_in[6];
    const int*   gtl      = (const int*)d_in[7];
    const unsigned char* gtv = (const unsigned char*)d_in[8];  // numpy bool = 1 byte

    const int C = 80;
    long long ncls = (long long)in_sizes[0];
    int P = in_sizes[4] / 2;
    int B = (int)(ncls / ((long long)P * C));
    int G = in_sizes[7] / B;
    if (G > MAXG) G = MAXG;

    float* ws = (float*)d_ws;
    const int NA = 1024;
    int gx = (P + 255) / 256;
    int NB = gx * B;
    float* partsA = ws;
    float* partsB = ws + NA;

    k_cls_softplus<<<NA, 256, 0, stream>>>(pred_cls, ncls, partsA);
    dim3 g2((unsigned)gx, (unsigned)B);
    k_assign<<<g2, 256, 0, stream>>>(pred_cls, pred_obj, dec, points, strides,
                                     gtb, gtl, gtv, P, G, C, partsB, NB);
    k_final<<<1, 256, 0, stream>>>(partsA, NA, partsB, NB, (float*)d_out);
}